// SelfAttentiveSpanExtractor_64355789964054
// MI455X (gfx1250) — compile-verified
//
#include <hip/hip_runtime.h>

#define BB 8
#define SS 512
#define DD 768
#define NN 512
#define MAXW 16
#define TINYV 1e-13f

typedef float v2f __attribute__((ext_vector_type(2)));
typedef float v8f __attribute__((ext_vector_type(8)));

// reduce within each 16-lane half of a wave32 (xor masks 1,2,4,8)
__device__ __forceinline__ float redmax16(float v) {
    v = fmaxf(v, __shfl_xor(v, 1, 32));
    v = fmaxf(v, __shfl_xor(v, 2, 32));
    v = fmaxf(v, __shfl_xor(v, 4, 32));
    v = fmaxf(v, __shfl_xor(v, 8, 32));
    return v;
}
__device__ __forceinline__ float redsum16(float v) {
    v += __shfl_xor(v, 1, 32);
    v += __shfl_xor(v, 2, 32);
    v += __shfl_xor(v, 4, 32);
    v += __shfl_xor(v, 8, 32);
    return v;
}

// ---------------------------------------------------------------------------
// Kernel 1: logits[b,s] = dot(seq[b,s,:], att_w) + att_b   (B*S rows of D)
// One wave per row; float4 streaming loads; full-wave shuffle reduction.
// ---------------------------------------------------------------------------
__global__ void span_logits_kernel(const float* __restrict__ seq,
                                   const float* __restrict__ attw,
                                   const float* __restrict__ attb,
                                   float* __restrict__ logits) {
    const int tid  = threadIdx.x;
    const int lane = tid & 31;
    const int row  = blockIdx.x * 8 + (tid >> 5);   // 8 waves per block

    const float4* rp = (const float4*)(seq + (size_t)row * DD);
    const float4* wp = (const float4*)attw;

    float s = 0.f;
#pragma unroll
    for (int k = 0; k < DD / 128; ++k) {            // 768/4 = 192 float4 / 32 lanes = 6
        float4 x = rp[lane + 32 * k];
        float4 w = wp[lane + 32 * k];
        s += x.x * w.x + x.y * w.y + x.z * w.z + x.w * w.w;
    }
    s += __shfl_xor(s, 16, 32);
    s += __shfl_xor(s, 8, 32);
    s += __shfl_xor(s, 4, 32);
    s += __shfl_xor(s, 2, 32);
    s += __shfl_xor(s, 1, 32);
    if (lane == 0) logits[row] = s + attb[0];
}

// ---------------------------------------------------------------------------
// Kernel 2: per span -> masked softmax over 16 logits, then
//           out[span, :] = attn(1x16) * emb(16x768) via V_WMMA_F32_16X16X4_F32
// One wave per span. K = span width (4 chained K=4 WMMAs),
// N = 16-wide d-tile, M redundant. Two d-chunks per loop iteration so the
// store is a fully coalesced 128B write across all 32 lanes.
// ---------------------------------------------------------------------------
__global__ void span_extract_kernel(const float* __restrict__ seq,
                                    const int*   __restrict__ spans,
                                    const float* __restrict__ logits,
                                    float*       __restrict__ out) {
    const int tid  = threadIdx.x;
    const int lane = tid & 31;
    const int L    = lane & 15;   // position within half-wave
    const int h    = lane >> 4;   // which half-wave

    const int span  = blockIdx.x * 8 + (tid >> 5);
    const int b     = span >> 9;                    // / NN
    const int start = spans[span * 2 + 0];
    const int end   = spans[span * 2 + 1];          // inclusive
    const int width = end - start;                  // in [0, 15]

    // ---- masked softmax over the 16 span logits (reference semantics:
    //      softmax(logits*mask) * mask, renormalized with +TINY) ----
    const float m  = (L <= width) ? 1.f : 0.f;
    const float l  = logits[b * SS + start + L] * m;
    const float mx = redmax16(l);
    const float e  = expf(l - mx);
    const float s1 = redsum16(e);
    const float a  = (e / s1) * m;
    const float s2 = redsum16(a);
    const float attn = a / (s2 + TINYV);            // lane L holds attn[L] (both halves)

    // ---- A operands (independent of d): A[m,k] = attn[4j+k] for all m.
    // ISA 16x4 f32 A layout: lanes 0-15 -> K=0 (vgpr0), K=1 (vgpr1);
    //                        lanes 16-31 -> K=2, K=3.
    v2f A[4];
#pragma unroll
    for (int j = 0; j < 4; ++j) {
        int w0 = 4 * j + 2 * h;
        A[j].x = __shfl(attn, w0, 32);
        A[j].y = __shfl(attn, w0 + 1, 32);
    }

    // ---- gathered row base offsets for the B tiles (masked w -> row `start`,
    //      contribution is exactly zero since attn=0 there) ----
    size_t rowoff[4][2];
    const size_t bbase = (size_t)b * SS * DD;
#pragma unroll
    for (int j = 0; j < 4; ++j) {
        int w0 = 4 * j + 2 * h;
        int w1 = w0 + 1;
        int t0 = start + ((w0 <= width) ? w0 : 0);
        int t1 = start + ((w1 <= width) ? w1 : 0);
        rowoff[j][0] = bbase + (size_t)t0 * DD;
        rowoff[j][1] = bbase + (size_t)t1 * DD;
    }

    float* outp = out + (size_t)span * DD;

#pragma unroll 1
    for (int d0 = 0; d0 < DD; d0 += 32) {
        v8f c0 = {0.f, 0.f, 0.f, 0.f, 0.f, 0.f, 0.f, 0.f};
        v8f c1 = {0.f, 0.f, 0.f, 0.f, 0.f, 0.f, 0.f, 0.f};
#pragma unroll
        for (int j = 0; j < 4; ++j) {
            // B[k,n] = emb[w=4j+k][d0+n]; lanes 0-15 carry k=0/1, 16-31 carry k=2/3
            v2f b0, b1;
            b0.x = seq[rowoff[j][0] + d0 + L];
            b0.y = seq[rowoff[j][1] + d0 + L];
            b1.x = seq[rowoff[j][0] + d0 + 16 + L];
            b1.y = seq[rowoff[j][1] + d0 + 16 + L];
            c0 = __builtin_amdgcn_wmma_f32_16x16x4_f32(false, A[j], false, b0,
                                                       (short)0, c0, false, false);
            c1 = __builtin_amdgcn_wmma_f32_16x16x4_f32(false, A[j], false, b1,
                                                       (short)0, c1, false, false);
        }
        // D vgpr0: lane n (0-15) = out[d0 + n] (chunk0); lanes 16-31 same for chunk1
        outp[d0 + lane] = h ? c1[0] : c0[0];
    }
}

extern "C" void kernel_launch(void* const* d_in, const int* in_sizes, int n_in,
                              void* d_out, int out_size, void* d_ws, size_t ws_size,
                              hipStream_t stream) {
    const float* seq   = (const float*)d_in[0];   // (B,S,D) f32
    const int*   spans = (const int*)d_in[1];     // (B,N,2) i32
    const float* attw  = (const float*)d_in[2];   // (D,)    f32
    const float* attb  = (const float*)d_in[3];   // (1,)    f32
    float*       outp  = (float*)d_out;           // (B,N,D) f32
    float*       logit = (float*)d_ws;            // B*S floats = 16 KB scratch

    // Pass 1: token logits (streams seq once, 12.6 MB)
    span_logits_kernel<<<(BB * SS) / 8, 256, 0, stream>>>(seq, attw, attb, logit);
    // Pass 2: masked softmax + WMMA weighted sum (L2-resident gathers)
    span_extract_kernel<<<(BB * NN) / 8, 256, 0, stream>>>(seq, spans, logit, outp);
}